// lstm_42185168781425
// MI455X (gfx1250) — compile-verified
//
#include <hip/hip_runtime.h>
#include <hip/hip_bf16.h>

// Problem constants from the reference.
#define B_ 4096
#define T_ 1024
#define I_ 6
#define H_ 2
#define L_ 5
#define G_ 8   // 4*H packed gates (i,f,g,o)

typedef __attribute__((ext_vector_type(2))) float v2f;
typedef __attribute__((ext_vector_type(8))) float v8f;

// ---------------------------------------------------------------------------
// Kernel 1: layer-0 input projection for the single live batch element.
//   pre0[t, g] = sum_k x_last[t, k] * Wih0[g, k] + bih0[g] + bhh0[g]
// Implemented as V_WMMA_F32_16X16X4_F32: M=16 timesteps per wave,
// K=6 padded to 8 (two k=4 steps), N=8 gates padded to 16.
//
// ISA lane layouts (wave32):
//   A 16x4:  lane L: m=L&15; VGPR j holds K = 2*(L>>4) + j
//   B 4x16:  lane L: n=L&15; VGPR j holds K = 2*(L>>4) + j   (row-striped)
//   C 16x16: lane L: n=L&15; VGPR r holds M = 8*(L>>4) + r
// ---------------------------------------------------------------------------
__global__ __launch_bounds__(32) void pre0_wmma_kernel(
    const float* __restrict__ x_last,  // [T_, I_]  (x + (B_-1)*T_*I_)
    const float* __restrict__ Wih0,    // [G_, I_]
    const float* __restrict__ bih0,    // [G_] (layer-0 row of bih)
    const float* __restrict__ bhh0,    // [G_] (layer-0 row of bhh)
    float* __restrict__ pre0)          // [T_, G_]
{
    const int lane = threadIdx.x & 31;
    const int half = lane >> 4;   // 0 or 1
    const int idx  = lane & 15;
    const int t0   = blockIdx.x * 16;

    // ----- A matrix: x rows (m = idx = timestep-in-tile) -----
    const int xrow = (t0 + idx) * I_;
    v2f a0, a1;
    a0.x = x_last[xrow + 2 * half + 0];              // K = 0..3, always < 6
    a0.y = x_last[xrow + 2 * half + 1];
    // second k-step covers K = 4..7; K=6,7 are zero padding
    a1.x = (half == 0) ? x_last[xrow + 4] : 0.0f;
    a1.y = (half == 0) ? x_last[xrow + 5] : 0.0f;

    // ----- B matrix: Wih0^T (n = idx = gate), rows n>=8 are zero padding ----
    const int nn = idx & (G_ - 1);                   // clamped safe address
    const bool nvalid = (idx < G_);
    v2f b0, b1;
    {
        const float w0 = Wih0[nn * I_ + 2 * half + 0];
        const float w1 = Wih0[nn * I_ + 2 * half + 1];
        b0.x = nvalid ? w0 : 0.0f;
        b0.y = nvalid ? w1 : 0.0f;
        const float w4 = Wih0[nn * I_ + 4];
        const float w5 = Wih0[nn * I_ + 5];
        b1.x = (nvalid && half == 0) ? w4 : 0.0f;
        b1.y = (nvalid && half == 0) ? w5 : 0.0f;
    }

    // ----- D = A*B + C over two K=4 steps -----
    v8f c = {};
    c = __builtin_amdgcn_wmma_f32_16x16x4_f32(false, a0, false, b0,
                                              (short)0, c, false, false);
    c = __builtin_amdgcn_wmma_f32_16x16x4_f32(false, a1, false, b1,
                                              (short)0, c, false, false);

    // ----- store valid 16x8 block with biases folded in -----
    if (nvalid) {
        const float bias = bih0[idx] + bhh0[idx];
#pragma unroll
        for (int r = 0; r < 8; ++r) {
            const int t = t0 + 8 * half + r;
            pre0[t * G_ + idx] = c[r] + bias;
        }
    }
}

// ---------------------------------------------------------------------------
// Kernel 2: 5-wave pipelined LSTM scan over the single live sequence.
// Wave l owns layer l; at global tick `tick` it processes t = tick - l.
// h sequences are exchanged through LDS; one block barrier per tick.
// Lanes 0..7 each own one packed gate; __shfl broadcasts i/f/g/o so every
// lane redundantly carries (c0,c1,h0,h1).
// ---------------------------------------------------------------------------
__device__ __forceinline__ float sigf(float x) {
    return 1.0f / (1.0f + expf(-x));
}

__global__ __launch_bounds__(L_ * 32) void lstm_scan_kernel(
    const float* __restrict__ pre0,  // [T_, G_]  (biases already included)
    const float* __restrict__ Wih,   // [L_-1, G_, H_]  layers 1..4
    const float* __restrict__ Whh,   // [L_,   G_, H_]
    const float* __restrict__ bih,   // [L_, G_]
    const float* __restrict__ bhh,   // [L_, G_]
    float* __restrict__ out)         // [H_]
{
    __shared__ float hbuf[L_][T_][H_];  // 40 KB of the WGP's 320 KB LDS

    const int tid  = threadIdx.x;
    const int wv   = tid >> 5;   // layer id 0..4 (wave32)
    const int lane = tid & 31;

    const bool gl  = (lane < G_);
    const int  gl8 = lane & (G_ - 1);  // clamped safe address

    // Per-gate recurrent weights (lanes 0..7).
    const float whh0 = gl ? Whh[(wv * G_ + gl8) * H_ + 0] : 0.0f;
    const float whh1 = gl ? Whh[(wv * G_ + gl8) * H_ + 1] : 0.0f;
    // Per-gate input weights + bias for layers >= 1.
    float wih0w = 0.0f, wih1w = 0.0f, bias = 0.0f;
    if (wv > 0) {
        const float a = Wih[((wv - 1) * G_ + gl8) * H_ + 0];
        const float b = Wih[((wv - 1) * G_ + gl8) * H_ + 1];
        const float bb = bih[wv * G_ + gl8] + bhh[wv * G_ + gl8];
        wih0w = gl ? a : 0.0f;
        wih1w = gl ? b : 0.0f;
        bias  = gl ? bb : 0.0f;
    }

    float h0 = 0.0f, h1 = 0.0f, c0 = 0.0f, c1 = 0.0f;

    for (int tick = 0; tick < T_ + L_ - 1; ++tick) {
        __syncthreads();  // uniform trip count -> legal barrier
        const int t = tick - wv;
        if (t >= 0 && t < T_) {
            float pre;
            if (wv == 0) {
                pre = gl ? pre0[t * G_ + gl8] : 0.0f;  // bias already folded
            } else {
                const float in0 = hbuf[wv - 1][t][0];
                const float in1 = hbuf[wv - 1][t][1];
                pre = bias + wih0w * in0 + wih1w * in1;
            }
            pre += whh0 * h0 + whh1 * h1;

            // Gather packed gates (PyTorch order i,f,g,o; H=2).
            const float gi0 = __shfl(pre, 0, 32), gi1 = __shfl(pre, 1, 32);
            const float gf0 = __shfl(pre, 2, 32), gf1 = __shfl(pre, 3, 32);
            const float gg0 = __shfl(pre, 4, 32), gg1 = __shfl(pre, 5, 32);
            const float go0 = __shfl(pre, 6, 32), go1 = __shfl(pre, 7, 32);

            c0 = sigf(gf0) * c0 + sigf(gi0) * tanhf(gg0);
            c1 = sigf(gf1) * c1 + sigf(gi1) * tanhf(gg1);
            h0 = sigf(go0) * tanhf(c0);
            h1 = sigf(go1) * tanhf(c1);

            if (lane == 0) {
                hbuf[wv][t][0] = h0;
                hbuf[wv][t][1] = h1;
            }
        }
    }

    // Final: softmax over the 2-element hidden of layer 4 at t = T-1.
    if (wv == L_ - 1 && lane == 0) {
        const float mx = fmaxf(h0, h1);
        const float e0 = expf(h0 - mx);
        const float e1 = expf(h1 - mx);
        const float inv = 1.0f / (e0 + e1);
        out[0] = e0 * inv;
        out[1] = e1 * inv;
    }
}

// ---------------------------------------------------------------------------
// Launch: inputs are (x, hidden, Wih0, Wih, Whh, bih, bhh); hidden unused.
// d_ws holds pre0 (T_*G_ floats = 32 KB), fully rewritten every call.
// ---------------------------------------------------------------------------
extern "C" void kernel_launch(void* const* d_in, const int* in_sizes, int n_in,
                              void* d_out, int out_size, void* d_ws, size_t ws_size,
                              hipStream_t stream) {
    const float* x    = (const float*)d_in[0];
    const float* Wih0 = (const float*)d_in[2];
    const float* Wih  = (const float*)d_in[3];
    const float* Whh  = (const float*)d_in[4];
    const float* bih  = (const float*)d_in[5];
    const float* bhh  = (const float*)d_in[6];
    float* out  = (float*)d_out;
    float* pre0 = (float*)d_ws;

    const float* x_last = x + (size_t)(B_ - 1) * T_ * I_;  // only live batch row

    pre0_wmma_kernel<<<T_ / 16, 32, 0, stream>>>(x_last, Wih0, bih, bhh, pre0);
    lstm_scan_kernel<<<1, L_ * 32, 0, stream>>>(pre0, Wih, Whh, bih, bhh, out);
}